// LocalClustering_62904091018180
// MI455X (gfx1250) — compile-verified
//
#include <hip/hip_runtime.h>
#include <hip/hip_bf16.h>
#include <stdint.h>

// ---------------------------------------------------------------------------
// LocalClustering (HGCalML) for MI455X / gfx1250.
// Serial greedy scan kept in LDS (taken-bitmap) with CDNA5 async-to-LDS DMA
// prefetch of neighbour rows (global_load_async_to_lds_b128 + s_wait_asynccnt),
// 64-wide batched skip scanning (claims restricted to the proven-resident
// 32-entry window), and a hybrid global/LDS bitonic sort for the ordering.
// ---------------------------------------------------------------------------

#define TPB 256
#define DEPTH 48            // async ring depth (rows of 64 neighbours)
#define CHK  64             // entries checked per serial step (2 per lane)
#define CLW  32             // claim window: rows proven resident by wait 0x10
#define MAXV 200704         // bitmap capacity (multiple of 32) >= V
#define BMW  (MAXV / 32)    // bitmap words in LDS: 6272 words = 25 KB

// ---------------- phase 0: init outputs / scratch --------------------------
__global__ void init_out_kernel(int* __restrict__ sel, int* __restrict__ rs_out,
                                int* __restrict__ is_centre, int V, int R) {
    int i = blockIdx.x * blockDim.x + threadIdx.x;
    if (i < V) { sel[i] = -1; is_centre[i] = 0; }
    if (i < R) rs_out[i] = 0;
}

// ---------------- phase 1: sort keys  (seg asc, hier desc, idx asc) --------
__global__ void init_keys_kernel(const float* __restrict__ hier,
                                 const int* __restrict__ rsplits, int R,
                                 unsigned long long* __restrict__ keys,
                                 int V, int npad) {
    int i = blockIdx.x * blockDim.x + threadIdx.x;
    if (i >= npad) return;
    if (i >= V) { keys[i] = ~0ull; return; }
    int seg = 0;
    for (int j = 1; j < R; ++j) seg += (i >= rsplits[j]) ? 1 : 0;
    unsigned fb = __float_as_uint(hier[i]);  // hier in [0,1): bits monotonic
    unsigned flip = ~fb;                     // descending hier
    keys[i] = ((unsigned long long)(unsigned)seg << 53)
            | ((unsigned long long)flip << 21)
            | (unsigned long long)(unsigned)i; // stable tie-break (idx asc)
}

__global__ void bitonic_stage_kernel(unsigned long long* __restrict__ keys,
                                     int j, int k, int n) {
    int i = blockIdx.x * blockDim.x + threadIdx.x;
    if (i >= n) return;
    int ixj = i ^ j;
    if (ixj > i) {
        unsigned long long a = keys[i], b = keys[ixj];
        bool up = ((i & k) == 0);
        if ((a > b) == up) { keys[i] = b; keys[ixj] = a; }
    }
}

// all steps j = jstart .. 1 of stage k, within a 2048-element LDS tile
__global__ __launch_bounds__(1024)
void bitonic_local_kernel(unsigned long long* __restrict__ keys,
                          int k, int jstart) {
    __shared__ unsigned long long sh[2048];
    int t = threadIdx.x;
    int base = blockIdx.x * 2048;
    sh[t]        = keys[base + t];
    sh[t + 1024] = keys[base + t + 1024];
    __syncthreads();
    for (int j = jstart; j >= 1; j >>= 1) {
        int i = ((t & ~(j - 1)) << 1) | (t & (j - 1));  // pair leader (bit j==0)
        unsigned long long a = sh[i], b = sh[i ^ j];
        bool up = (((base + i) & k) == 0);
        if ((a > b) == up) { sh[i] = b; sh[i ^ j] = a; }
        __syncthreads();
    }
    keys[base + t]        = sh[t];
    keys[base + t + 1024] = sh[t + 1024];
}

__global__ void extract_order_kernel(const unsigned long long* __restrict__ keys,
                                     int* __restrict__ order, int V) {
    int i = blockIdx.x * blockDim.x + threadIdx.x;
    if (i < V) order[i] = (int)(keys[i] & 0x1FFFFFull);
}

// ---------------- phase 2: serial greedy scan (one wave32) -----------------
// 64-wide batched skip-scan over the taken bitmap; neighbour rows streamed
// DEPTH ahead into an LDS ring by CDNA5 async DMA loads (ASYNCcnt pipeline).
// Claims happen only at first-free index f < CLW, where residency is proven.
__global__ __launch_bounds__(32)
void greedy_scan_kernel(const int* __restrict__ order,
                        const int* __restrict__ neighs,   // [V*64]
                        int* __restrict__ centre_of,
                        int* __restrict__ is_centre,
                        int V) {
    __shared__ unsigned bitmap[BMW];
    __shared__ int ring[DEPTH * 64];
    __shared__ int ff;

    const int lane = threadIdx.x;           // wave32: 0..31
    for (int w = lane; w < BMW; w += 32) bitmap[w] = 0u;
    __syncthreads();

    // low 32 bits of the flat address == LDS byte offset (shared aperture has
    // zero low bits), which is what the async-load VDST expects.
    unsigned ring_off = (unsigned)(unsigned long long)(&ring[0]);
    unsigned long long nb_base = (unsigned long long)(uintptr_t)neighs;

    // prime the ring: DEPTH rows in flight
    int head = (V < DEPTH) ? V : DEPTH;
    for (int p = 0; p < head; ++p) {
        int row = order[p];
        if (lane < 16) {
            unsigned l = ring_off + (unsigned)((p % DEPTH) * 256 + lane * 16);
            unsigned g = (unsigned)row * 256u + (unsigned)(lane * 16);
            asm volatile("global_load_async_to_lds_b128 %0, %1, %2"
                         :: "v"(l), "v"(g), "s"(nb_base) : "memory");
        }
    }

    int pos = 0;
    while (pos < V) {
        __builtin_prefetch(order + pos + 1024, 0, 0);

        // invariant: head == min(pos+DEPTH, V); after the wait, rows
        // [pos, pos+CLW) are resident (loads retire in order; outstanding<=16)
        if (head < V) {
            asm volatile("s_wait_asynccnt 0x10" ::: "memory"); // DEPTH-CLW=16
        } else {
            asm volatile("s_wait_asynccnt 0x0" ::: "memory");
        }

        // check 64 order entries (2 per lane) against current bitmap state
        int p0 = pos + lane;
        int p1 = pos + 32 + lane;
        int o0 = (p0 < V) ? order[p0] : 0;
        int o1 = (p1 < V) ? order[p1] : 0;
        int cand = CHK;
        if (p0 < V) {
            unsigned w = bitmap[(unsigned)o0 >> 5];
            if (((w >> (o0 & 31)) & 1u) == 0u) cand = lane;
        }
        if (cand == CHK && p1 < V) {
            unsigned w = bitmap[(unsigned)o1 >> 5];
            if (((w >> (o1 & 31)) & 1u) == 0u) cand = 32 + lane;
        }
        if (lane == 0) ff = CHK;
        __syncthreads();
        if (cand < CHK) atomicMin(&ff, cand);
        __syncthreads();
        int f = ff;

        int adv;
        if (f < CLW) {
            // first free entry == next centre of the sequential algorithm
            int cpos = pos + f;
            int idx = __shfl(o0, f, 32);
            int slot = cpos % DEPTH;
            int n0 = ring[slot * 64 + lane];
            int n1 = ring[slot * 64 + 32 + lane];
            if (n0 >= 0 && n0 < V) {
                unsigned bit = 1u << (n0 & 31);
                unsigned old = atomicOr(&bitmap[(unsigned)n0 >> 5], bit);
                if ((old & bit) == 0u) centre_of[n0] = idx;
            }
            if (n1 >= 0 && n1 < V) {
                unsigned bit = 1u << (n1 & 31);
                unsigned old = atomicOr(&bitmap[(unsigned)n1 >> 5], bit);
                if ((old & bit) == 0u) centre_of[n1] = idx;
            }
            if (lane == 0) is_centre[idx] = 1;
            adv = f + 1;                   // re-check entries after the centre
        } else if (f < CHK) {
            adv = CLW;                     // free entry beyond resident window:
                                           // advance; next step claims it
        } else {
            adv = (V - pos < CHK) ? (V - pos) : CHK;      // skip whole batch
        }

        // refill the consumed window: issue async loads for [head, new_head)
        int new_head = pos + adv + DEPTH;
        if (new_head > V) new_head = V;
        if (new_head > head) {
            int a0 = head + lane, a1 = head + 32 + lane;
            int r0 = (a0 < V) ? order[a0] : 0;
            int r1 = (a1 < V) ? order[a1] : 0;
            // ring reads of the just-consumed slots must retire before the
            // async DMA overwrites those same slots (q%DEPTH aliases them)
            asm volatile("s_wait_dscnt 0x0" ::: "memory");
            for (int q = head; q < new_head; ++q) {
                int d = q - head;
                int row = (d < 32) ? __shfl(r0, d, 32) : __shfl(r1, d - 32, 32);
                if (lane < 16) {
                    unsigned l = ring_off + (unsigned)((q % DEPTH) * 256 + lane * 16);
                    unsigned g = (unsigned)row * 256u + (unsigned)(lane * 16);
                    asm volatile("global_load_async_to_lds_b128 %0, %1, %2"
                                 :: "v"(l), "v"(g), "s"(nb_base) : "memory");
                }
            }
            head = new_head;
        }
        pos += adv;
    }
}

// ---------------- phase 3: cluster ids via exclusive scan ------------------
__global__ void build_flags_kernel(const int* __restrict__ order,
                                   const int* __restrict__ is_centre,
                                   int* __restrict__ flag, int V) {
    int i = blockIdx.x * blockDim.x + threadIdx.x;
    if (i < V) flag[i] = is_centre[order[i]];
}

__global__ void scan_block_kernel(const int* __restrict__ flag,
                                  int* __restrict__ scanarr,
                                  int* __restrict__ bsums, int n) {
    __shared__ int tmp[256];
    int t = threadIdx.x;
    int g = blockIdx.x * 256 + t;
    int v = (g < n) ? flag[g] : 0;
    tmp[t] = v;
    __syncthreads();
    for (int off = 1; off < 256; off <<= 1) {
        int x = (t >= off) ? tmp[t - off] : 0;
        __syncthreads();
        tmp[t] += x;
        __syncthreads();
    }
    if (g < n) scanarr[g] = tmp[t] - v;            // exclusive
    if (t == 255) bsums[blockIdx.x] = tmp[t];      // block total
}

__global__ void scan_sums_kernel(int* __restrict__ bsums, int nb,
                                 int* __restrict__ nclu_out) {
    __shared__ int tmp[1024];
    int t = threadIdx.x;
    int v = (t < nb) ? bsums[t] : 0;
    tmp[t] = v;
    __syncthreads();
    for (int off = 1; off < 1024; off <<= 1) {
        int x = (t >= off) ? tmp[t - off] : 0;
        __syncthreads();
        tmp[t] += x;
        __syncthreads();
    }
    if (t < nb) bsums[t] = tmp[t] - v;             // exclusive
    if (t == 1023) *nclu_out = tmp[t];             // total clusters
}

__global__ void finalize_centres_kernel(const int* __restrict__ flag,
                                        const int* __restrict__ scanarr,
                                        const int* __restrict__ bsums,
                                        const int* __restrict__ order,
                                        int* __restrict__ centre_id,
                                        int* __restrict__ sel_out, int V) {
    int i = blockIdx.x * blockDim.x + threadIdx.x;
    if (i >= V) return;
    int id = scanarr[i] + bsums[i >> 8];
    if (flag[i]) {
        int c = order[i];
        centre_id[c] = id;
        sel_out[id] = c;
    }
}

__global__ void write_ggather_kernel(const int* __restrict__ centre_of,
                                     const int* __restrict__ centre_id,
                                     int* __restrict__ gg, int V) {
    int i = blockIdx.x * blockDim.x + threadIdx.x;
    if (i >= V) return;
    int c = centre_of[i];
    gg[i] = (c >= 0 && c < V) ? centre_id[c] : -1;
}

__global__ void count_rs_kernel(const int* __restrict__ is_centre,
                                const int* __restrict__ rsplits,
                                int* __restrict__ rs_out, int V, int R) {
    int i = blockIdx.x * blockDim.x + threadIdx.x;
    if (i >= V) return;
    if (is_centre[i]) {
        for (int j = 0; j < R; ++j)
            if (i < rsplits[j]) atomicAdd(&rs_out[j], 1);
    }
}

// ---------------------------------------------------------------------------
static inline size_t align256(size_t x) { return (x + 255) & ~(size_t)255; }

extern "C" void kernel_launch(void* const* d_in, const int* in_sizes, int n_in,
                              void* d_out, int out_size, void* d_ws, size_t ws_size,
                              hipStream_t stream) {
    (void)n_in; (void)out_size; (void)ws_size;
    const int*   neighs  = (const int*)d_in[0];
    const float* hier    = (const float*)d_in[1];
    const int*   rsplits = (const int*)d_in[2];
    int V = in_sizes[1];     // hier has V elements
    int R = in_sizes[2];     // row_splits length

    unsigned npad = 1;
    while (npad < (unsigned)V) npad <<= 1;

    // workspace carve-up (keys region recycled for flag+scanarr after sort)
    char* ws = (char*)d_ws;
    unsigned long long* keys = (unsigned long long*)ws;
    size_t off = align256((size_t)npad * 8);
    int* order     = (int*)(ws + off); off += align256((size_t)V * 4);
    int* centre_of = (int*)(ws + off); off += align256((size_t)V * 4);
    int* is_centre = (int*)(ws + off); off += align256((size_t)V * 4);
    int* centre_id = (int*)(ws + off); off += align256((size_t)V * 4);
    int* bsums     = (int*)(ws + off); off += align256(1024 * 4);
    int* flag      = (int*)ws;                       // reuse keys[0 .. )
    int* scanarr   = (int*)(ws + (size_t)npad * 4);  // reuse keys upper half

    // output layout: sel[V], rs[R], ggather[V], nclu[1]
    int* out_i = (int*)d_out;
    int* sel_o = out_i;
    int* rs_o  = out_i + V;
    int* gg_o  = out_i + V + R;
    int* nclu_o = out_i + 2 * V + R;

    int gV = (V + TPB - 1) / TPB;
    int gN = ((int)npad + TPB - 1) / TPB;
    int nb = (V + 255) / 256;

    init_out_kernel<<<gV, TPB, 0, stream>>>(sel_o, rs_o, is_centre, V, R);
    init_keys_kernel<<<gN, TPB, 0, stream>>>(hier, rsplits, R, keys, V, (int)npad);

    if (npad >= 2048) {
        for (unsigned k = 2; k <= npad; k <<= 1) {
            for (unsigned j = k >> 1; j >= 2048; j >>= 1)
                bitonic_stage_kernel<<<gN, TPB, 0, stream>>>(keys, (int)j, (int)k, (int)npad);
            unsigned jstart = ((k >> 1) < 1024u) ? (k >> 1) : 1024u;
            bitonic_local_kernel<<<(int)(npad / 2048), 1024, 0, stream>>>(keys, (int)k, (int)jstart);
        }
    } else {
        for (unsigned k = 2; k <= npad; k <<= 1)
            for (unsigned j = k >> 1; j >= 1; j >>= 1)
                bitonic_stage_kernel<<<gN, TPB, 0, stream>>>(keys, (int)j, (int)k, (int)npad);
    }

    extract_order_kernel<<<gV, TPB, 0, stream>>>(keys, order, V);

    greedy_scan_kernel<<<1, 32, 0, stream>>>(order, neighs, centre_of, is_centre, V);

    build_flags_kernel<<<gV, TPB, 0, stream>>>(order, is_centre, flag, V);
    scan_block_kernel<<<nb, 256, 0, stream>>>(flag, scanarr, bsums, V);
    scan_sums_kernel<<<1, 1024, 0, stream>>>(bsums, nb, nclu_o);
    finalize_centres_kernel<<<gV, TPB, 0, stream>>>(flag, scanarr, bsums, order,
                                                    centre_id, sel_o, V);
    write_ggather_kernel<<<gV, TPB, 0, stream>>>(centre_of, centre_id, gg_o, V);
    count_rs_kernel<<<gV, TPB, 0, stream>>>(is_centre, rsplits, rs_o, V, R);
}